// MRConv2d_11922829214263
// MI455X (gfx1250) — compile-verified
//
#include <hip/hip_runtime.h>

typedef float v2f __attribute__((ext_vector_type(2)));
typedef float v8f __attribute__((ext_vector_type(8)));

#define B_SZ    8
#define C_IN    192
#define N_PTS   3136
#define K_NB    16
#define GROUPS  4
#define C_OUT   192
#define CPG     48      // original channels per group
#define FAN     96      // 2*CPG (interleaved xs/rel)
#define TN      64      // points per block (4 waves x 16)
#define XG_STR  80      // padded LDS stride for xg tile
#define W_STR   97      // padded LDS stride for W tile

__global__ __launch_bounds__(128) void mrconv2d_wmma_kernel(
    const float* __restrict__ x,      // (B, C, N, 1)
    const int*   __restrict__ edge,   // (2, B, N, K)
    const float* __restrict__ W,      // (GROUPS, CPG, FAN)
    const float* __restrict__ bias,   // (C_OUT)
    float*       __restrict__ out)    // (B, C_OUT, N, 1)
{
    __shared__ float xg[FAN * XG_STR];     // 30720 B
    __shared__ float wl[CPG * W_STR];      // 18624 B
    __shared__ int   idx_j[TN * K_NB];     // 4096 B
    __shared__ int   idx_i[TN * K_NB];     // 4096 B

    const int tid    = threadIdx.x;
    const int n_base = blockIdx.x * TN;    // 0..48 tiles, 49*64 == 3136
    const int g      = blockIdx.y;
    const int b      = blockIdx.z;

    // ---- stage neighbor indices for this tile into LDS ----
    const long ebase0 = (long)b * (N_PTS * K_NB) + (long)n_base * K_NB;           // edge[0]
    const long ebase1 = (long)B_SZ * (N_PTS * K_NB) + ebase0;                     // edge[1]
    for (int t = tid; t < TN * K_NB; t += 128) {
        idx_j[t] = edge[ebase0 + t];
        idx_i[t] = edge[ebase1 + t];
    }
    // ---- stage this group's weights into LDS (padded rows) ----
    const float* Wg = W + (long)g * (CPG * FAN);
    for (int t = tid; t < CPG * FAN; t += 128) {
        int o = t / FAN, i = t - o * FAN;
        wl[o * W_STR + i] = Wg[t];
    }
    __syncthreads();

    // ---- phase 1: build interleaved xg[96][64] tile (xs even rows, rel odd rows) ----
    const float* xb = x + (long)b * (C_IN * N_PTS);
    for (int p = tid; p < CPG * TN; p += 128) {
        const int c_local = p >> 6;        // p / 64
        const int nl      = p & 63;
        const int c       = g * CPG + c_local;
        const float* xrow = xb + (long)c * N_PTS;

        const float xv = xrow[n_base + nl];
        float rel = -3.402823466e38f;
        #pragma unroll
        for (int k = 0; k < K_NB; ++k) {
            const int jj = idx_j[nl * K_NB + k];
            const int ii = idx_i[nl * K_NB + k];
            rel = fmaxf(rel, xrow[jj] - xrow[ii]);
        }
        xg[(2 * c_local + 0) * XG_STR + nl] = xv;
        xg[(2 * c_local + 1) * XG_STR + nl] = rel;
    }
    __syncthreads();

    // ---- phase 2: per-wave 48x16 = W(48x96) * xg(96x16) via v_wmma_f32_16x16x4_f32 ----
    const int wave  = tid >> 5;
    const int lane  = tid & 31;
    const int lhalf = lane >> 4;           // 0: K pair {0,1}, 1: K pair {2,3}
    const int l16   = lane & 15;
    const int n0    = wave * 16;           // this wave's 16-point column slab

    v8f acc0 = {}; v8f acc1 = {}; v8f acc2 = {};

    for (int k0 = 0; k0 < FAN; k0 += 4) {
        const int kb = k0 + lhalf * 2;

        // B fragment: 4x16, row K striped across lanes; VGPR0 = K {0|2}, VGPR1 = K {1|3}
        v2f bf;
        bf.x = xg[(kb + 0) * XG_STR + n0 + l16];
        bf.y = xg[(kb + 1) * XG_STR + n0 + l16];

        // A fragments: 16x4 rows of W for each of the 3 output 16-row tiles
        v2f a0, a1, a2;
        a0.x = wl[(0 * 16 + l16) * W_STR + kb + 0];
        a0.y = wl[(0 * 16 + l16) * W_STR + kb + 1];
        a1.x = wl[(1 * 16 + l16) * W_STR + kb + 0];
        a1.y = wl[(1 * 16 + l16) * W_STR + kb + 1];
        a2.x = wl[(2 * 16 + l16) * W_STR + kb + 0];
        a2.y = wl[(2 * 16 + l16) * W_STR + kb + 1];

        acc0 = __builtin_amdgcn_wmma_f32_16x16x4_f32(false, a0, false, bf, (short)0, acc0, false, false);
        acc1 = __builtin_amdgcn_wmma_f32_16x16x4_f32(false, a1, false, bf, (short)0, acc1, false, false);
        acc2 = __builtin_amdgcn_wmma_f32_16x16x4_f32(false, a2, false, bf, (short)0, acc2, false, false);
    }

    // ---- epilogue: bias + relu + store (D VGPR v -> rows M=v / M=v+8) ----
    const int n = n_base + n0 + l16;
    float* outb = out + (long)b * (C_OUT * N_PTS);
    #pragma unroll
    for (int v = 0; v < 8; ++v) {
        {
            const int ch = g * CPG + 0 * 16 + v + lhalf * 8;
            outb[(long)ch * N_PTS + n] = fmaxf(acc0[v] + bias[ch], 0.0f);
        }
        {
            const int ch = g * CPG + 1 * 16 + v + lhalf * 8;
            outb[(long)ch * N_PTS + n] = fmaxf(acc1[v] + bias[ch], 0.0f);
        }
        {
            const int ch = g * CPG + 2 * 16 + v + lhalf * 8;
            outb[(long)ch * N_PTS + n] = fmaxf(acc2[v] + bias[ch], 0.0f);
        }
    }
}

extern "C" void kernel_launch(void* const* d_in, const int* in_sizes, int n_in,
                              void* d_out, int out_size, void* d_ws, size_t ws_size,
                              hipStream_t stream) {
    const float* x    = (const float*)d_in[0];
    const int*   edge = (const int*)d_in[1];
    const float* W    = (const float*)d_in[2];
    const float* bias = (const float*)d_in[3];
    float*       out  = (float*)d_out;

    dim3 grid(N_PTS / TN, GROUPS, B_SZ);   // 49 x 4 x 8
    dim3 block(128);
    mrconv2d_wmma_kernel<<<grid, block, 0, stream>>>(x, edge, W, bias, out);
}